// BRGCN_10093173145881
// MI455X (gfx1250) — compile-verified
//
#include <hip/hip_runtime.h>
#include <hip/hip_bf16.h>
#include <math.h>

typedef __attribute__((ext_vector_type(2))) float v2f;
typedef __attribute__((ext_vector_type(8))) float v8f;

#define NN 50000
#define EE 640000
#define RR 8
#define HH 4
#define CC 32
#define KDIM 128          // IN == H*C == 128
#define NEG_SLOPE 0.2f
#define EPSV 1e-16f
#define ENC_NEG_INF 0x007FFFFFu   // enc_f32(-inf)

// ---------- monotonic float<->uint encoding for atomic max ----------
__device__ __forceinline__ unsigned enc_f32(float f) {
  unsigned u = __float_as_uint(f);
  return (u & 0x80000000u) ? ~u : (u | 0x80000000u);
}
__device__ __forceinline__ float dec_f32(unsigned u) {
  return __uint_as_float((u & 0x80000000u) ? (u & 0x7FFFFFFFu) : ~u);
}

// ---------- generic fill ----------
__global__ void fill_u32(unsigned* __restrict__ p, unsigned val, long long n) {
  long long i = (long long)blockIdx.x * blockDim.x + threadIdx.x;
  if (i < n) p[i] = val;
}

// ---------- f32 WMMA GEMM: C[M,Nc] = A[M,128] @ B[128,Nc] ----------
// One wave computes one 16x16 output tile. total_tiles = (M/16)*(Nc/16).
__global__ void gemm_wmma_f32(const float* __restrict__ A, const float* __restrict__ B,
                              float* __restrict__ Cmat, int Nc, int total_tiles, int ntiles) {
  int wid = (int)(((long long)blockIdx.x * blockDim.x + threadIdx.x) >> 5);
  if (wid >= total_tiles) return;                  // wave-uniform
  int lane = threadIdx.x & 31;
  int mt = wid / ntiles;
  int nt = wid - mt * ntiles;
  int col  = lane & 15;                            // row for A, col for B/D
  int half = lane >> 4;                            // lane-half selects K pair / M+8
  const float* Arow = A + (size_t)(mt * 16 + col) * KDIM;
  const float* Bcol = B + nt * 16 + col;
  v8f acc = {};
  for (int kk = 0; kk < KDIM; kk += 4) {
    int kb = kk + half * 2;
    v2f a, b;
    a[0] = Arow[kb];
    a[1] = Arow[kb + 1];
    b[0] = Bcol[(size_t)kb * Nc];
    b[1] = Bcol[(size_t)(kb + 1) * Nc];
    acc = __builtin_amdgcn_wmma_f32_16x16x4_f32(false, a, false, b, (short)0, acc, false, false);
  }
  float* Crow = Cmat + (size_t)(mt * 16 + half * 8) * Nc + nt * 16 + col;
#pragma unroll
  for (int g = 0; g < 8; ++g) Crow[(size_t)g * Nc] = acc[g];
}

// ---------- q/k/v GEMMs with fused z = agg[r] + sn ----------
// grid = (3125, R), block = 32 (one wave). Computes a 16-row slab of q,k,v[r].
__global__ void qkv_wmma(const float* __restrict__ agg, const float* __restrict__ sn,
                         const float* __restrict__ Wq, const float* __restrict__ Wk,
                         const float* __restrict__ Wv,
                         float* __restrict__ q, float* __restrict__ k, float* __restrict__ v) {
  int mt = blockIdx.x;
  int r  = blockIdx.y;
  int lane = threadIdx.x & 31;
  int col  = lane & 15;
  int half = lane >> 4;
  const float* Az = agg + ((size_t)r * NN + (size_t)mt * 16 + col) * KDIM;
  const float* Sz = sn  + ((size_t)mt * 16 + col) * KDIM;
  const float* Bq = Wq + (size_t)r * KDIM * CC + col;
  const float* Bk = Wk + (size_t)r * KDIM * CC + col;
  const float* Bv = Wv + (size_t)r * KDIM * CC + col;
  v8f aq0 = {}, aq1 = {}, ak0 = {}, ak1 = {}, av0 = {}, av1 = {};
  for (int kk = 0; kk < KDIM; kk += 4) {
    int kb = kk + half * 2;
    v2f a, b;
    a[0] = Az[kb]     + Sz[kb];
    a[1] = Az[kb + 1] + Sz[kb + 1];
    size_t o0 = (size_t)kb * CC, o1 = (size_t)(kb + 1) * CC;
    b[0] = Bq[o0];      b[1] = Bq[o1];
    aq0 = __builtin_amdgcn_wmma_f32_16x16x4_f32(false, a, false, b, (short)0, aq0, false, false);
    b[0] = Bq[o0 + 16]; b[1] = Bq[o1 + 16];
    aq1 = __builtin_amdgcn_wmma_f32_16x16x4_f32(false, a, false, b, (short)0, aq1, false, false);
    b[0] = Bk[o0];      b[1] = Bk[o1];
    ak0 = __builtin_amdgcn_wmma_f32_16x16x4_f32(false, a, false, b, (short)0, ak0, false, false);
    b[0] = Bk[o0 + 16]; b[1] = Bk[o1 + 16];
    ak1 = __builtin_amdgcn_wmma_f32_16x16x4_f32(false, a, false, b, (short)0, ak1, false, false);
    b[0] = Bv[o0];      b[1] = Bv[o1];
    av0 = __builtin_amdgcn_wmma_f32_16x16x4_f32(false, a, false, b, (short)0, av0, false, false);
    b[0] = Bv[o0 + 16]; b[1] = Bv[o1 + 16];
    av1 = __builtin_amdgcn_wmma_f32_16x16x4_f32(false, a, false, b, (short)0, av1, false, false);
  }
  size_t base = ((size_t)r * NN + (size_t)mt * 16 + half * 8) * CC + col;
#pragma unroll
  for (int g = 0; g < 8; ++g) {
    q[base + (size_t)g * CC]      = aq0[g];
    q[base + (size_t)g * CC + 16] = aq1[g];
    k[base + (size_t)g * CC]      = ak0[g];
    k[base + (size_t)g * CC + 16] = ak1[g];
    v[base + (size_t)g * CC]      = av0[g];
    v[base + (size_t)g * CC + 16] = av1[g];
  }
}

// ---------- per-node attention dot precompute ----------
// ai[n,r,h] = sum_c att[r,h,c]   * h_i[n,h,c]
// aj[n,r,h] = sum_c att[r,h,C+c] * h_j[n,h,c]
__global__ void node_dots(const float* __restrict__ hi, const float* __restrict__ hj,
                          const float* __restrict__ natt,
                          float* __restrict__ ai, float* __restrict__ aj) {
  int n = blockIdx.x * blockDim.x + threadIdx.x;
  if (n >= NN) return;
  const float4* xi = (const float4*)(hi + (size_t)n * KDIM);
  const float4* xj = (const float4*)(hj + (size_t)n * KDIM);
#pragma unroll
  for (int h = 0; h < HH; ++h) {
    float4 vi[8], vj[8];
#pragma unroll
    for (int c4 = 0; c4 < 8; ++c4) { vi[c4] = xi[h * 8 + c4]; vj[c4] = xj[h * 8 + c4]; }
#pragma unroll
    for (int r = 0; r < RR; ++r) {
      const float4* at = (const float4*)(natt + ((size_t)r * HH + h) * 2 * CC);
      float di = 0.f, dj = 0.f;
#pragma unroll
      for (int c4 = 0; c4 < 8; ++c4) {
        float4 a0 = at[c4];       // att[:, :C]
        float4 a1 = at[8 + c4];   // att[:, C:]
        di += a0.x * vi[c4].x + a0.y * vi[c4].y + a0.z * vi[c4].z + a0.w * vi[c4].w;
        dj += a1.x * vj[c4].x + a1.y * vj[c4].y + a1.z * vj[c4].z + a1.w * vj[c4].w;
      }
      ai[((size_t)n * RR + r) * HH + h] = di;
      aj[((size_t)n * RR + r) * HH + h] = dj;
    }
  }
}

// ---------- per-edge attention logits + segment max (32B gather per edge) ----------
__global__ void edge_alpha(const float* __restrict__ ai, const float* __restrict__ aj,
                           const int* __restrict__ ei, const int* __restrict__ et,
                           float* __restrict__ alpha, unsigned* __restrict__ amax) {
  int e = blockIdx.x * blockDim.x + threadIdx.x;
  if (e >= EE) return;
  int s = ei[e];
  int d = ei[EE + e];
  int r = et[e];
  float4 vi = *(const float4*)(ai + ((size_t)d * RR + r) * HH);
  float4 vj = *(const float4*)(aj + ((size_t)s * RR + r) * HH);
  float a4[HH] = { vi.x + vj.x, vi.y + vj.y, vi.z + vj.z, vi.w + vj.w };
  size_t segb = ((size_t)r * NN + d) * HH;
#pragma unroll
  for (int h = 0; h < HH; ++h) {
    float acc = a4[h];
    acc = acc > 0.f ? acc : acc * NEG_SLOPE;   // leaky relu
    alpha[(size_t)e * HH + h] = acc;
    atomicMax(&amax[segb + h], enc_f32(acc));
  }
}

// ---------- exp(alpha - amax) + segment sum ----------
__global__ void edge_exp(const int* __restrict__ ei, const int* __restrict__ et,
                         const unsigned* __restrict__ amax,
                         float* __restrict__ alpha, float* __restrict__ denom) {
  int i = blockIdx.x * blockDim.x + threadIdx.x;   // over EE*HH
  if (i >= EE * HH) return;
  int e = i >> 2;
  int h = i & 3;
  int d = ei[EE + e];
  int r = et[e];
  size_t seg = ((size_t)r * NN + d) * HH + h;
  unsigned mu = amax[seg];
  float m = (mu == ENC_NEG_INF) ? 0.f : dec_f32(mu);
  float ex = expf(alpha[i] - m);
  alpha[i] = ex;                                   // reuse buffer
  atomicAdd(&denom[seg], ex);
}

// ---------- normalized message scatter into agg[R,N,128] ----------
__global__ void edge_scatter(const int* __restrict__ ei, const int* __restrict__ et,
                             const float* __restrict__ hj, const float* __restrict__ alpha,
                             const float* __restrict__ denom, float* __restrict__ agg) {
  long long t = (long long)blockIdx.x * blockDim.x + threadIdx.x;  // over EE*32 float4-groups
  if (t >= (long long)EE * 32) return;
  int e = (int)(t >> 5);
  int g = (int)(t & 31);       // float4 group within the 128-wide row
  int h = g >> 3;
  int s = ei[e];
  int d = ei[EE + e];
  int r = et[e];
  size_t seg = ((size_t)r * NN + d) * HH + h;
  float a = alpha[(size_t)e * HH + h] / (denom[seg] + EPSV);
  float4 vj = ((const float4*)(hj + (size_t)s * KDIM))[g];
  float* dstp = agg + ((size_t)r * NN + d) * KDIM + (size_t)g * 4;
  atomicAdd(dstp + 0, a * vj.x);
  atomicAdd(dstp + 1, a * vj.y);
  atomicAdd(dstp + 2, a * vj.z);
  atomicAdd(dstp + 3, a * vj.w);
}

// ---------- relation-level attention + output ----------
__global__ void relattn_final(const float* __restrict__ q, const float* __restrict__ k,
                              const float* __restrict__ v, const float* __restrict__ self_term,
                              const float* __restrict__ wrel, float* __restrict__ out) {
  int n = blockIdx.x * blockDim.x + threadIdx.x;
  if (n >= NN) return;
  float st[CC], outacc[CC];
  const float4* stp = (const float4*)(self_term + (size_t)n * CC);
#pragma unroll
  for (int c4 = 0; c4 < 8; ++c4) {
    float4 tv = stp[c4];
    st[c4 * 4 + 0] = tv.x; st[c4 * 4 + 1] = tv.y;
    st[c4 * 4 + 2] = tv.z; st[c4 * 4 + 3] = tv.w;
  }
#pragma unroll
  for (int c = 0; c < CC; ++c) outacc[c] = 0.f;

  for (int r = 0; r < RR; ++r) {
    float qr[CC];
    const float4* qp = (const float4*)(q + ((size_t)r * NN + n) * CC);
#pragma unroll
    for (int c4 = 0; c4 < 8; ++c4) {
      float4 tv = qp[c4];
      qr[c4 * 4 + 0] = tv.x; qr[c4 * 4 + 1] = tv.y;
      qr[c4 * 4 + 2] = tv.z; qr[c4 * 4 + 3] = tv.w;
    }
    float psi[RR];
#pragma unroll
    for (int s = 0; s < RR; ++s) {
      const float4* kp = (const float4*)(k + ((size_t)s * NN + n) * CC);
      float dot = 0.f;
#pragma unroll
      for (int c4 = 0; c4 < 8; ++c4) {
        float4 kv = kp[c4];
        dot += qr[c4 * 4 + 0] * kv.x + qr[c4 * 4 + 1] * kv.y
             + qr[c4 * 4 + 2] * kv.z + qr[c4 * 4 + 3] * kv.w;
      }
      psi[s] = dot;
    }
    float mx = psi[0];
#pragma unroll
    for (int s = 1; s < RR; ++s) mx = fmaxf(mx, psi[s]);
    float sum = 0.f;
#pragma unroll
    for (int s = 0; s < RR; ++s) { psi[s] = expf(psi[s] - mx); sum += psi[s]; }
    float inv = 1.f / sum;

    float delta[CC];
#pragma unroll
    for (int c = 0; c < CC; ++c) delta[c] = 0.f;
#pragma unroll
    for (int s = 0; s < RR; ++s) {
      float p = psi[s] * inv;
      const float4* vp = (const float4*)(v + ((size_t)s * NN + n) * CC);
#pragma unroll
      for (int c4 = 0; c4 < 8; ++c4) {
        float4 vv = vp[c4];
        delta[c4 * 4 + 0] += p * vv.x; delta[c4 * 4 + 1] += p * vv.y;
        delta[c4 * 4 + 2] += p * vv.z; delta[c4 * 4 + 3] += p * vv.w;
      }
    }
    float dsum = 0.f;
#pragma unroll
    for (int c = 0; c < CC; ++c) dsum += delta[c];
    float mask = (dsum != 0.f) ? 1.f : 0.f;
    float wr = wrel[r];
#pragma unroll
    for (int c = 0; c < CC; ++c) outacc[c] += wr * (delta[c] + st[c] * mask);
  }
  float4* op = (float4*)(out + (size_t)n * CC);
#pragma unroll
  for (int c4 = 0; c4 < 8; ++c4) {
    float4 tv;
    tv.x = outacc[c4 * 4 + 0]; tv.y = outacc[c4 * 4 + 1];
    tv.z = outacc[c4 * 4 + 2]; tv.w = outacc[c4 * 4 + 3];
    op[c4] = tv;
  }
}

extern "C" void kernel_launch(void* const* d_in, const int* in_sizes, int n_in,
                              void* d_out, int out_size, void* d_ws, size_t ws_size,
                              hipStream_t stream) {
  (void)in_sizes; (void)n_in; (void)out_size; (void)ws_size;
  const float* x     = (const float*)d_in[0];
  const int*   ei    = (const int*)d_in[1];   // [2,E] flat: src then dst
  const int*   et    = (const int*)d_in[2];
  const float* Wj    = (const float*)d_in[3];
  const float* Wi    = (const float*)d_in[4];
  const float* natt  = (const float*)d_in[5];
  const float* Wq    = (const float*)d_in[6];
  const float* Wk    = (const float*)d_in[7];
  const float* Wv    = (const float*)d_in[8];
  const float* Wself = (const float*)d_in[9];
  const float* Wsn   = (const float*)d_in[10];
  const float* Wrel  = (const float*)d_in[11];
  float* out = (float*)d_out;

  float* ws = (float*)d_ws;
  size_t off = 0;
  float* hj        = ws + off; off += (size_t)NN * KDIM;      // 6.4M
  float* hi        = ws + off; off += (size_t)NN * KDIM;      // 6.4M
  float* sn        = ws + off; off += (size_t)NN * KDIM;      // 6.4M
  float* self_term = ws + off; off += (size_t)NN * CC;        // 1.6M
  float* alpha     = ws + off; off += (size_t)EE * HH;        // 2.56M
  unsigned* amax   = (unsigned*)(ws + off); off += (size_t)RR * NN * HH;  // 1.6M
  float* denom     = ws + off; off += (size_t)RR * NN * HH;   // 1.6M
  float* ai        = ws + off; off += (size_t)NN * RR * HH;   // 1.6M
  float* aj        = ws + off; off += (size_t)NN * RR * HH;   // 1.6M
  float* agg       = ws + off; off += (size_t)RR * NN * KDIM; // 51.2M
  float* q         = ws + off; off += (size_t)RR * NN * CC;   // 12.8M
  float* k         = ws + off; off += (size_t)RR * NN * CC;   // 12.8M
  float* v         = ws + off; off += (size_t)RR * NN * CC;   // 12.8M

  // init: agg=0, denom=0, amax=enc(-inf)
  {
    long long n_agg = (long long)RR * NN * KDIM;
    fill_u32<<<(unsigned)((n_agg + 255) / 256), 256, 0, stream>>>((unsigned*)agg, 0u, n_agg);
    long long n_seg = (long long)RR * NN * HH;
    fill_u32<<<(unsigned)((n_seg + 255) / 256), 256, 0, stream>>>((unsigned*)denom, 0u, n_seg);
    fill_u32<<<(unsigned)((n_seg + 255) / 256), 256, 0, stream>>>(amax, ENC_NEG_INF, n_seg);
  }

  const int mtiles = NN / 16;  // 3125 exactly
  // x @ {Wj, Wi, W_self_node}: Nc=128 (8 col-tiles)
  {
    int tiles = mtiles * 8;
    int blocks = (tiles * 32 + 255) / 256;
    gemm_wmma_f32<<<blocks, 256, 0, stream>>>(x, Wj,  hj, KDIM, tiles, 8);
    gemm_wmma_f32<<<blocks, 256, 0, stream>>>(x, Wi,  hi, KDIM, tiles, 8);
    gemm_wmma_f32<<<blocks, 256, 0, stream>>>(x, Wsn, sn, KDIM, tiles, 8);
  }
  // x @ W_self: Nc=32 (2 col-tiles)
  {
    int tiles = mtiles * 2;
    int blocks = (tiles * 32 + 255) / 256;
    gemm_wmma_f32<<<blocks, 256, 0, stream>>>(x, Wself, self_term, CC, tiles, 2);
  }

  node_dots<<<(NN + 255) / 256, 256, 0, stream>>>(hi, hj, natt, ai, aj);
  edge_alpha<<<(EE + 255) / 256, 256, 0, stream>>>(ai, aj, ei, et, alpha, amax);
  edge_exp<<<(EE * HH + 255) / 256, 256, 0, stream>>>(ei, et, amax, alpha, denom);
  {
    long long t = (long long)EE * 32;
    edge_scatter<<<(unsigned)((t + 255) / 256), 256, 0, stream>>>(ei, et, hj, alpha, denom, agg);
  }

  qkv_wmma<<<dim3(mtiles, RR), 32, 0, stream>>>(agg, sn, Wq, Wk, Wv, q, k, v);

  relattn_final<<<(NN + 255) / 256, 256, 0, stream>>>(q, k, v, self_term, Wrel, out);
}